// WeightedHausdorffDistance_68393059222122
// MI455X (gfx1250) — compile-verified
//
#include <hip/hip_runtime.h>
#include <hip/hip_bf16.h>

#define HH 256
#define WW 256
#define NPIX 65536
#define BATCH 8
#define MPTS 100
#define MPAD 112           // 7 tiles of 16
#define NTILES 7
#define EPSC 1e-6f
#define MAXD 362.038671967511f   // sqrt(256^2+256^2)
#define ACC_SIZE (2*BATCH + BATCH*MPAD)   // [term1 num][n_est][s sums]

typedef __attribute__((ext_vector_type(2))) float v2f;
typedef __attribute__((ext_vector_type(8))) float v8f;

// raw hardware transcendentals: single v_sqrt_f32 / v_log_f32 / v_exp_f32,
// no libm IEEE-fixup expansion. pow(x,p) = exp2(p * log2(x)).
__device__ __forceinline__ float hw_sqrt(float x)  { return __builtin_amdgcn_sqrtf(x); }
__device__ __forceinline__ float hw_powr(float x, float p) {
    return __builtin_amdgcn_exp2f(p * __builtin_amdgcn_logf(x));
}

__global__ void whd_zero(float* __restrict__ acc) {
    int i = threadIdx.x;
    if (i < ACC_SIZE) acc[i] = 0.0f;
}

// Pairwise-distance core:  d^2 = ||x||^2 + ||y||^2 - 2 x.y  as a K=4 WMMA.
// A row (pixel):  [ yi, xi, yi^2+xi^2, 1 ]
// B col (point):  [ -2*ym, -2*xm, 1, ym^2+xm^2 ]
__global__ __launch_bounds__(256) void whd_main(
        const float* __restrict__ prob, const float* __restrict__ gt,
        const float* __restrict__ osz,  float* __restrict__ acc) {
    __shared__ float s_gy[MPAD], s_gx[MPAD], s_sum[MPAD], s_p[128];
    __shared__ float s_t1, s_ne;

    const int b    = blockIdx.x >> 9;          // 512 blocks per batch
    const int pblk = (blockIdx.x & 511) * 128; // 128 pixels per block
    const int tid  = threadIdx.x;

    const float nrmy = osz[2*b+0] * (1.0f / (float)HH);
    const float nrmx = osz[2*b+1] * (1.0f / (float)WW);

    if (tid < MPAD) {
        s_sum[tid] = 0.0f;
        if (tid < MPTS) {
            s_gy[tid] = nrmy * gt[(b*MPTS + tid)*2 + 0];
            s_gx[tid] = nrmx * gt[(b*MPTS + tid)*2 + 1];
        } else {
            s_gy[tid] = 1.0e7f;   // pad: never wins min, vanishes in ^-9 sum
            s_gx[tid] = 1.0e7f;
        }
    }
    if (tid < 128) s_p[tid] = prob[b*NPIX + pblk + tid];
    if (tid == 0) { s_t1 = 0.0f; s_ne = 0.0f; }
    __syncthreads();

    const int wv   = tid >> 5;
    const int lane = tid & 31;
    const int r    = lane & 15;   // row within A-half / col within B-half
    const int half = lane >> 4;

    // A fragment: 16 pixels (rows) of this wave
    const int pix  = pblk + wv*16 + r;
    const float yi = nrmy * (float)(pix >> 8);
    const float xi = nrmx * (float)(pix & 255);
    v2f afrag;
    afrag.x = half ? (yi*yi + xi*xi) : yi;   // K=2 : K=0
    afrag.y = half ? 1.0f            : xi;   // K=3 : K=1

    // p for my 8 output rows (C layout: vgpr j -> row j + 8*half)
    float pv[8];
    #pragma unroll
    for (int j = 0; j < 8; ++j) pv[j] = s_p[wv*16 + 8*half + j];

    float vmin[8];
    #pragma unroll
    for (int j = 0; j < 8; ++j) vmin[j] = 3.4e38f;

    #pragma unroll
    for (int mt = 0; mt < NTILES; ++mt) {
        const int m = mt*16 + r;
        const float gy = s_gy[m], gx = s_gx[m];
        v2f bfrag;
        bfrag.x = half ? 1.0f              : (-2.0f * gy);  // K=2 : K=0
        bfrag.y = half ? (gy*gy + gx*gx)   : (-2.0f * gx);  // K=3 : K=1

        v8f cacc = {};
        cacc = __builtin_amdgcn_wmma_f32_16x16x4_f32(
            /*neg_a=*/false, afrag, /*neg_b=*/false, bfrag,
            /*c_mod=*/(short)0, cacc, /*reuse_a=*/false, /*reuse_b=*/false);

        float colsum = 0.0f;
        #pragma unroll
        for (int j = 0; j < 8; ++j) {
            float d = hw_sqrt(cacc[j] + EPSC);          // sqrt(dist^2 + eps)
            vmin[j] = fminf(vmin[j], d);
            float p  = pv[j];
            float wd = (1.0f - p) * MAXD + p * d;
            colsum += hw_powr(wd + EPSC, -9.0f);        // (w+eps)^-9
        }
        // combine the two half-wave partial sums for column m, then LDS reduce
        colsum += __shfl_xor(colsum, 16, 32);
        if (half == 0) atomicAdd(&s_sum[m], colsum);
    }

    // per-row min over the 16 columns held across lanes of each half
    #pragma unroll
    for (int j = 0; j < 8; ++j) {
        float v = vmin[j];
        v = fminf(v, __shfl_xor(v, 1, 32));
        v = fminf(v, __shfl_xor(v, 2, 32));
        v = fminf(v, __shfl_xor(v, 4, 32));
        v = fminf(v, __shfl_xor(v, 8, 32));
        vmin[j] = v;
    }
    if (r == 0) {  // lanes 0 and 16: rows 0-7 and 8-15
        float t1 = 0.0f, ne = 0.0f;
        #pragma unroll
        for (int j = 0; j < 8; ++j) { t1 += pv[j] * vmin[j]; ne += pv[j]; }
        atomicAdd(&s_t1, t1);
        atomicAdd(&s_ne, ne);
    }
    __syncthreads();

    if (tid < MPAD) atomicAdd(&acc[2*BATCH + b*MPAD + tid], s_sum[tid]);
    if (tid == 0) {
        atomicAdd(&acc[b], s_t1);
        atomicAdd(&acc[BATCH + b], s_ne);
    }
}

__global__ __launch_bounds__(256) void whd_final(
        const float* __restrict__ acc, const float* __restrict__ pc,
        const float* __restrict__ tc,  float* __restrict__ out) {
    __shared__ float red[256];
    const int tid = threadIdx.x;
    float part = 0.0f;
    for (int i = tid; i < BATCH*MPTS; i += 256) {
        int b = i / MPTS, m = i % MPTS;
        float s = acc[2*BATCH + b*MPAD + m] * (1.0f / (float)NPIX); // mean over n
        part += hw_powr(s, -1.0f/9.0f);                              // ^(1/P)
    }
    red[tid] = part;
    __syncthreads();
    for (int off = 128; off > 0; off >>= 1) {
        if (tid < off) red[tid] += red[tid + off];
        __syncthreads();
    }
    if (tid == 0) {
        float term2 = red[0] / (float)(BATCH * MPTS);
        float term1 = 0.0f;
        for (int b = 0; b < BATCH; ++b)
            term1 += acc[b] / (acc[BATCH + b] + EPSC);
        term1 /= (float)BATCH;
        float sl = 0.0f;
        for (int b = 0; b < BATCH; ++b) {
            float d = pc[b] - tc[b];
            float ad = fabsf(d);
            sl += (ad < 1.0f) ? 0.5f*d*d : (ad - 0.5f);
        }
        sl /= (float)BATCH;
        out[0] = sl + term1 + term2;
    }
}

extern "C" void kernel_launch(void* const* d_in, const int* in_sizes, int n_in,
                              void* d_out, int out_size, void* d_ws, size_t ws_size,
                              hipStream_t stream) {
    const float* prob = (const float*)d_in[0];  // [8,256,256]
    const float* gt   = (const float*)d_in[1];  // [8,100,2]
    const float* osz  = (const float*)d_in[2];  // [8,2]
    const float* pc   = (const float*)d_in[3];  // [8]
    const float* tc   = (const float*)d_in[4];  // [8]
    float* out = (float*)d_out;
    float* acc = (float*)d_ws;                  // 912 floats

    whd_zero<<<1, 1024, 0, stream>>>(acc);
    whd_main<<<BATCH * (NPIX / 128), 256, 0, stream>>>(prob, gt, osz, acc);
    whd_final<<<1, 256, 0, stream>>>(acc, pc, tc, out);
}